// GNNEncoderCrossAttn_13451837571852
// MI455X (gfx1250) — compile-verified
//
#include <hip/hip_runtime.h>
#include <hip/hip_bf16.h>

// ---------------- problem constants (match reference) ----------------
#define NN   100000
#define EE   1600000
#define ENE  (EE + NN)        // edges + self loops
#define HID  128
#define NH   4
#define CH   32
#define EDIM 6
#define NLAY 4
#define INF_ 8
#define OUTF 256

typedef __attribute__((ext_vector_type(2))) float v2f;
typedef __attribute__((ext_vector_type(8))) float v8f;

// monotonic float<->uint encoding for atomicMax on signed floats
__device__ __forceinline__ unsigned enc_f(float f) {
    unsigned u = __float_as_uint(f);
    return (u & 0x80000000u) ? ~u : (u | 0x80000000u);
}
__device__ __forceinline__ float dec_f(unsigned u) {
    return (u & 0x80000000u) ? __uint_as_float(u ^ 0x80000000u) : __uint_as_float(~u);
}
__device__ __forceinline__ float elu_f(float x)   { return x > 0.f ? x : expm1f(x); }
__device__ __forceinline__ float leaky_f(float x) { return x > 0.f ? x : 0.2f * x; }

// ---------------- init kernels ----------------
__global__ void k_zero_f(float* __restrict__ p, int n) {
    int i = blockIdx.x * blockDim.x + threadIdx.x;
    if (i < n) p[i] = 0.f;
}
__global__ void k_zero_u(unsigned* __restrict__ p, int n) {
    int i = blockIdx.x * blockDim.x + threadIdx.x;
    if (i < n) p[i] = 0u;   // dec_f never read for empty segments (self-loops guarantee >=1)
}

// ---------------- self-loop attr: degree + segment sum ----------------
__global__ void k_deg_loop(const int* __restrict__ ei, const float* __restrict__ eattr,
                           float* __restrict__ deg, float* __restrict__ loopsum) {
    int e = blockIdx.x * blockDim.x + threadIdx.x;
    if (e >= EE) return;
    int dst = ei[EE + e];
    atomicAdd(&deg[dst], 1.f);
#pragma unroll
    for (int d = 0; d < EDIM; ++d)
        atomicAdd(&loopsum[(size_t)dst * EDIM + d], eattr[(size_t)e * EDIM + d]);
}
__global__ void k_loop_div(const float* __restrict__ loopsum, const float* __restrict__ deg,
                           float* __restrict__ loop_attr) {
    int i = blockIdx.x * blockDim.x + threadIdx.x;
    if (i >= NN * EDIM) return;
    loop_attr[i] = loopsum[i] / fmaxf(deg[i / EDIM], 1.f);
}

// ---------------- input projection: h = elu(x @ W_in + b_in) ----------------
__global__ void k_inproj(const float* __restrict__ x, const float* __restrict__ W,
                         const float* __restrict__ b, float* __restrict__ h) {
    int i = blockIdx.x * blockDim.x + threadIdx.x;   // over N*HID
    int n = i >> 7, c = i & 127;
    float acc = b[c];
#pragma unroll
    for (int d = 0; d < INF_; ++d) acc += x[(size_t)n * INF_ + d] * W[d * HID + c];
    h[i] = elu_f(acc);
}

// ---------------- fp32 WMMA GEMM: C[M x ncol] = A[M x 128] @ B[128 x ncol] + bias ----
// block = 256 threads (8 waves); wave w computes col-tile (blockIdx.y*8 + w);
// blockIdx.x = 16-row tile; A tile staged in LDS.
__global__ void k_gemm128(const float* __restrict__ A, const float* __restrict__ B,
                          const float* __restrict__ bias, float* __restrict__ C, int ncol) {
    __shared__ float As[16 * 128];
    const int tid = threadIdx.x;
    const int m0 = blockIdx.x * 16;
#pragma unroll
    for (int i = 0; i < 8; ++i) {
        int idx = tid + i * 256;
        As[idx] = A[(size_t)(m0 + (idx >> 7)) * 128 + (idx & 127)];
    }
    __syncthreads();

    const int wave = tid >> 5;
    const int lane = tid & 31;
    const int ct   = blockIdx.y * 8 + wave;
    const int col  = ct * 16 + (lane & 15);
    const int row  = lane & 15;           // A: lanes 0-15 -> M=row K={0,1}; 16-31 -> K={2,3}
    const int kb   = (lane >> 4) * 2;

    v8f c = {};
#pragma unroll 4
    for (int k0 = 0; k0 < 128; k0 += 4) {
        v2f a, b;
        a.x = As[row * 128 + k0 + kb];
        a.y = As[row * 128 + k0 + kb + 1];
        b.x = B[(size_t)(k0 + kb) * ncol + col];
        b.y = B[(size_t)(k0 + kb + 1) * ncol + col];
        c = __builtin_amdgcn_wmma_f32_16x16x4_f32(false, a, false, b, (short)0, c, false, false);
    }
    const float bv = bias[col];
    const int rbase = m0 + ((lane >> 4) * 8);
#pragma unroll
    for (int v = 0; v < 8; ++v)
        C[(size_t)(rbase + v) * ncol + col] = c[v] + bv;
}

// ---------------- per-(edge,head) attention score + segment max ----------------
// block = 256 = 64 edges x 4 heads; W_e (6x128) and att (128) staged in LDS.
__global__ void k_score(const int* __restrict__ ei, const float* __restrict__ eattr,
                        const float* __restrict__ loop_attr,
                        const float* __restrict__ xl, const float* __restrict__ xr,
                        const float* __restrict__ We, const float* __restrict__ att,
                        float* __restrict__ scorebuf, unsigned* __restrict__ smax) {
    __shared__ float s_att[HID];
    __shared__ float s_We[EDIM * HID];
    int tid = threadIdx.x;
    if (tid < HID) s_att[tid] = att[tid];
    for (int i = tid; i < EDIM * HID; i += 256) s_We[i] = We[i];
    __syncthreads();

    int e  = blockIdx.x * 64 + (tid >> 2);
    int hh = tid & 3;
    if (e >= ENE) return;

    int src, dst;
    float ea[EDIM];
    if (e < EE) {
        src = ei[e]; dst = ei[EE + e];
#pragma unroll
        for (int d = 0; d < EDIM; ++d) ea[d] = eattr[(size_t)e * EDIM + d];
    } else {
        src = dst = e - EE;
#pragma unroll
        for (int d = 0; d < EDIM; ++d) ea[d] = loop_attr[(size_t)src * EDIM + d];
    }
    const float* xlp = xl + (size_t)src * HID + hh * CH;
    const float* xrp = xr + (size_t)dst * HID + hh * CH;
    float sc = 0.f;
#pragma unroll
    for (int c = 0; c < CH; ++c) {
        float eac = 0.f;
#pragma unroll
        for (int d = 0; d < EDIM; ++d) eac += ea[d] * s_We[d * HID + hh * CH + c];
        sc += leaky_f(xlp[c] + xrp[c] + eac) * s_att[hh * CH + c];
    }
    scorebuf[(size_t)e * NH + hh] = sc;
    atomicMax(&smax[(size_t)dst * NH + hh], enc_f(sc));
}

// ---------------- exp(score - max) + segment sum ----------------
__global__ void k_expsum(const int* __restrict__ ei, float* __restrict__ scorebuf,
                         const unsigned* __restrict__ smax, float* __restrict__ denom) {
    int i = blockIdx.x * blockDim.x + threadIdx.x;
    if (i >= ENE * NH) return;
    int e = i >> 2, hh = i & 3;
    int dst = (e < EE) ? ei[EE + e] : (e - EE);
    float ex = __expf(scorebuf[i] - dec_f(smax[(size_t)dst * NH + hh]));
    scorebuf[i] = ex;
    atomicAdd(&denom[(size_t)dst * NH + hh], ex);
}

// ---------------- weighted scatter: out[dst] += alpha * xl[src] ----------------
// one block (128 threads) per edge; thread = channel
__global__ void k_scatter(const int* __restrict__ ei, const float* __restrict__ scorebuf,
                          const float* __restrict__ denom, const float* __restrict__ xl,
                          float* __restrict__ out) {
    int e = blockIdx.x;
    int c = threadIdx.x;
    int hh = c >> 5;
    int src, dst;
    if (e < EE) { src = ei[e]; dst = ei[EE + e]; }
    else        { src = dst = e - EE; }
    float al = scorebuf[(size_t)e * NH + hh] / denom[(size_t)dst * NH + hh];
    atomicAdd(&out[(size_t)dst * HID + c], al * xl[(size_t)src * HID + c]);
}

// ---------------- bias + residual + layernorm + elu (one wave32 per node) -------
__global__ void k_lnelu(const float* __restrict__ out, float* __restrict__ h,
                        const float* __restrict__ bias, const float* __restrict__ lnw,
                        const float* __restrict__ lnb) {
    int node = blockIdx.x * 8 + (threadIdx.x >> 5);
    int lane = threadIdx.x & 31;
    const float4* o4 = (const float4*)out;
    float4 o = o4[(size_t)node * 32 + lane];
    float4 r = ((const float4*)h)[(size_t)node * 32 + lane];
    float4 bi = ((const float4*)bias)[lane];
    float v0 = o.x + r.x + bi.x, v1 = o.y + r.y + bi.y;
    float v2 = o.z + r.z + bi.z, v3 = o.w + r.w + bi.w;
    float s  = v0 + v1 + v2 + v3;
    float sq = v0 * v0 + v1 * v1 + v2 * v2 + v3 * v3;
#pragma unroll
    for (int off = 16; off > 0; off >>= 1) {
        s  += __shfl_xor(s, off);
        sq += __shfl_xor(sq, off);
    }
    float mean = s * (1.f / HID);
    float var  = sq * (1.f / HID) - mean * mean;
    float inv  = rsqrtf(var + 1e-5f);
    float4 w = ((const float4*)lnw)[lane];
    float4 b = ((const float4*)lnb)[lane];
    float4 res;
    res.x = elu_f((v0 - mean) * inv * w.x + b.x);
    res.y = elu_f((v1 - mean) * inv * w.y + b.y);
    res.z = elu_f((v2 - mean) * inv * w.z + b.z);
    res.w = elu_f((v3 - mean) * inv * w.w + b.w);
    ((float4*)h)[(size_t)node * 32 + lane] = res;
}

static inline int cdiv(long long a, long long b) { return (int)((a + b - 1) / b); }

extern "C" void kernel_launch(void* const* d_in, const int* in_sizes, int n_in,
                              void* d_out, int out_size, void* d_ws, size_t ws_size,
                              hipStream_t stream) {
    const float* x      = (const float*)d_in[0];
    const int*   ei     = (const int*)  d_in[1];
    const float* eattr  = (const float*)d_in[2];
    const float* W_in   = (const float*)d_in[3];
    const float* b_in   = (const float*)d_in[4];
    const float* W_l    = (const float*)d_in[5];
    const float* b_l    = (const float*)d_in[6];
    const float* W_r    = (const float*)d_in[7];
    const float* b_r    = (const float*)d_in[8];
    const float* W_e    = (const float*)d_in[9];
    const float* att    = (const float*)d_in[10];
    const float* biasL  = (const float*)d_in[11];
    const float* ln_w   = (const float*)d_in[12];
    const float* ln_b   = (const float*)d_in[13];
    const float* W_out  = (const float*)d_in[14];
    const float* b_out  = (const float*)d_in[15];
    float* outp = (float*)d_out;

    // workspace layout (floats)
    float* ws = (float*)d_ws;
    const size_t NH128 = (size_t)NN * HID;           // 12.8M
    float*    h_buf     = ws;
    float*    xl_buf    = ws + NH128;
    float*    xr_buf    = ws + 2 * NH128;
    float*    out_buf   = ws + 3 * NH128;
    float*    score_buf = ws + 4 * NH128;            // ENE*NH = 6.8M
    float*    denom_buf = score_buf + (size_t)ENE * NH;
    unsigned* smax_buf  = (unsigned*)(denom_buf + (size_t)NN * NH);
    float*    deg_buf   = (float*)(smax_buf + (size_t)NN * NH);
    float*    loopsum   = deg_buf + NN;
    float*    loop_attr = loopsum + (size_t)NN * EDIM;

    // ---- self-loop edge attrs (fill_value='mean') ----
    k_zero_f<<<cdiv(NN, 256), 256, 0, stream>>>(deg_buf, NN);
    k_zero_f<<<cdiv(NN * EDIM, 256), 256, 0, stream>>>(loopsum, NN * EDIM);
    k_deg_loop<<<cdiv(EE, 256), 256, 0, stream>>>(ei, eattr, deg_buf, loopsum);
    k_loop_div<<<cdiv(NN * EDIM, 256), 256, 0, stream>>>(loopsum, deg_buf, loop_attr);

    // ---- input projection ----
    k_inproj<<<cdiv((long long)NN * HID, 256), 256, 0, stream>>>(x, W_in, b_in, h_buf);

    // ---- layers ----
    for (int i = 0; i < NLAY; ++i) {
        k_zero_f<<<cdiv((long long)NN * HID, 256), 256, 0, stream>>>(out_buf, NN * HID);
        k_zero_f<<<cdiv(NN * NH, 256), 256, 0, stream>>>(denom_buf, NN * NH);
        k_zero_u<<<cdiv(NN * NH, 256), 256, 0, stream>>>(smax_buf, NN * NH);

        dim3 ggrid(NN / 16, 1);
        k_gemm128<<<ggrid, 256, 0, stream>>>(h_buf, W_l + (size_t)i * HID * HID,
                                             b_l + i * HID, xl_buf, HID);
        k_gemm128<<<ggrid, 256, 0, stream>>>(h_buf, W_r + (size_t)i * HID * HID,
                                             b_r + i * HID, xr_buf, HID);

        k_score<<<cdiv(ENE, 64), 256, 0, stream>>>(ei, eattr, loop_attr, xl_buf, xr_buf,
                                                   W_e + (size_t)i * EDIM * HID,
                                                   att + (size_t)i * NH * CH,
                                                   score_buf, smax_buf);
        k_expsum<<<cdiv((long long)ENE * NH, 256), 256, 0, stream>>>(ei, score_buf, smax_buf,
                                                                     denom_buf);
        k_scatter<<<ENE, 128, 0, stream>>>(ei, score_buf, denom_buf, xl_buf, out_buf);
        k_lnelu<<<NN / 8, 256, 0, stream>>>(out_buf, h_buf, biasL + i * HID,
                                            ln_w + i * HID, ln_b + i * HID);
    }

    // ---- output projection: d_out = h @ W_out + b_out (N x 256) ----
    dim3 ogrid(NN / 16, OUTF / 128);
    k_gemm128<<<ogrid, 256, 0, stream>>>(h_buf, W_out, b_out, outp, OUTF);
}